// ChamferLoss_39548058862073
// MI455X (gfx1250) — compile-verified
//
#include <hip/hip_runtime.h>
#include <hip/hip_bf16.h>
#include <math.h>

// ---------------------------------------------------------------------------
// Chamfer loss via V_WMMA_F32_16X16X4_F32 (CDNA5 / gfx1250, wave32)
//
// dist2(p,q) = ||p||^2 - 2 p.q + ||q||^2, computed as a 16x16 tile per WMMA:
//   A row (16x4 f32)  = (-2p0, -2p1, -2p2, 1)        ("row" augmentation)
//   B col (4x16 f32)  = ( q0,   q1,   q2,  ||q||^2)  ("col" augmentation)
//   C     (16x16 f32) = ||p_m||^2 broadcast per row
//
// Symmetry trick: run the SAME row-min kernel twice with p/q swapped.
//   pass A: rows = y, cols = x  -> min over j per y point  (min2)
//   pass B: rows = x, cols = y  -> min over i per x point  (min1)
// Row-min is a pure per-register v_min accumulation (M fixed per (VGPR,lane)
// slot of the C/D layout) -> no shuffles, no atomics, no exec-mask branches
// in the hot loop. min over dist == sqrt(eps + min over dist^2).
// ---------------------------------------------------------------------------

typedef __attribute__((ext_vector_type(2))) float v2f;
typedef __attribute__((ext_vector_type(8))) float v8f;

#define NPTS 4096
#define BS   4
#define NTILE (NPTS / 16)          // 256 column tiles per row strip

// ---------------- Phase 0: build augmented forms ----------------
__global__ void __launch_bounds__(256)
chamfer_setup(const float* __restrict__ x, const float* __restrict__ y,
              float4* __restrict__ x_col, float4* __restrict__ x_row,
              float* __restrict__ nx,
              float4* __restrict__ y_col, float4* __restrict__ y_row,
              float* __restrict__ ny, int n) {
    int i = blockIdx.x * 256 + threadIdx.x;
    if (i >= n) return;
    float x0 = x[i * 3 + 0], x1 = x[i * 3 + 1], x2 = x[i * 3 + 2];
    float nxx = x0 * x0 + x1 * x1 + x2 * x2;
    x_col[i] = make_float4(x0, x1, x2, nxx);
    x_row[i] = make_float4(-2.0f * x0, -2.0f * x1, -2.0f * x2, 1.0f);
    nx[i] = nxx;
    float y0 = y[i * 3 + 0], y1 = y[i * 3 + 1], y2 = y[i * 3 + 2];
    float nyy = y0 * y0 + y1 * y1 + y2 * y2;
    y_col[i] = make_float4(y0, y1, y2, nyy);
    y_row[i] = make_float4(-2.0f * y0, -2.0f * y1, -2.0f * y2, 1.0f);
    ny[i] = nyy;
}

// ---------------- Phase 1: per-row min of dist^2 (run twice, roles swapped) --
// Grid: BS*32 blocks, 256 threads = 8 waves. Wave = one 16-row strip of the
// "row" point set vs all 4096 "col" points (col set resident in LDS, 64 KB).
__global__ void __launch_bounds__(256)
chamfer_minrows(const float4* __restrict__ colP,   // col-augmented opposing set
                const float4* __restrict__ rowP,   // row-augmented own set
                const float* __restrict__ nrm,     // ||own||^2
                float* __restrict__ rmin_out) {    // per-row min of dist^2
    __shared__ float4 qs[NPTS];        // 64 KB: col-augmented points, this batch

    const int b   = blockIdx.x >> 5;   // 32 blocks per batch
    const int grp = blockIdx.x & 31;
    const int tid = threadIdx.x;

    const float4* qb = colP + b * NPTS;
    for (int i = tid; i < NPTS; i += 256) qs[i] = qb[i];   // b128 -> ds b128
    __syncthreads();

    const int wave  = tid >> 5;
    const int lane  = tid & 31;
    const int half  = lane >> 4;       // 0: K=0,1 | 1: K=2,3 (A/B layout halves)
    const int l16   = lane & 15;
    const int strip = grp * 8 + wave;  // 0..255
    const int i0    = strip * 16;      // first row of this wave's strip

    // A matrix 16x4 f32 (2 VGPRs): lane L<16 holds (K0,K1) of row M=L,
    // lane L+16 holds (K2,K3) of row M=L.            (ISA 7.12.2)
    const float* prow = (const float*)(rowP + b * NPTS + i0 + l16);
    v2f a;
    a.x = prow[half * 2 + 0];
    a.y = prow[half * 2 + 1];

    // C matrix: C[v][lane] = ||p_{i0 + v + 8*half}||^2
    v8f c;
    const float* nb = nrm + b * NPTS + i0 + 8 * half;
#pragma unroll
    for (int v = 0; v < 8; ++v) c[v] = nb[v];

    v8f rmin;
#pragma unroll
    for (int v = 0; v < 8; ++v) rmin[v] = 3.0e38f;

    const float* qsf = (const float*)qs;
    const int laneoff = l16 * 4 + half * 2;

    // Double-buffered B tiles: prefetch (wrap-indexed, branch-free) so the
    // ds_load of tile jt+1 overlaps the WMMA + min of tile jt.
    v2f bnext;
    {
        const float* bp = qsf + laneoff;
        bnext.x = bp[0];
        bnext.y = bp[1];
    }
#pragma unroll 2
    for (int jt = 0; jt < NTILE; ++jt) {
        v2f bm = bnext;
        const float* bp = qsf + (((jt + 1) & (NTILE - 1)) * 64 + laneoff);
        bnext.x = bp[0];               // ds_load_b64 for next tile
        bnext.y = bp[1];

        // D = A*B + C : 16x16 tile of squared distances
        v8f d = __builtin_amdgcn_wmma_f32_16x16x4_f32(
            false, a, false, bm, (short)0, c, false, false);

        // running min over columns: M fixed per (v,lane) slot -> pure v_min
#pragma unroll
        for (int v = 0; v < 8; ++v) rmin[v] = fminf(rmin[v], d[v]);
    }

    // fold the 16 lanes of each half (each lane covered columns = l16 mod 16)
#pragma unroll
    for (int mask = 1; mask <= 8; mask <<= 1) {
#pragma unroll
        for (int v = 0; v < 8; ++v)
            rmin[v] = fminf(rmin[v], __shfl_xor(rmin[v], mask, 32));
    }
    if (l16 == 0) {
        // lane 0 -> rows i0..i0+7, lane 16 -> rows i0+8..i0+15 (exclusive writer)
        float* rw = rmin_out + b * NPTS + i0 + 8 * half;
#pragma unroll
        for (int v = 0; v < 8; ++v) rw[v] = fmaxf(rmin[v], 0.0f);
    }
}

// ---------------- Phase 2: sqrt(eps + min) and means ----------------
__global__ void __launch_bounds__(256)
chamfer_final(const float* __restrict__ min1, const float* __restrict__ min2,
              float* __restrict__ out, int n, float inv) {
    __shared__ float red[256];
    float s = 0.0f;
    for (int i = threadIdx.x; i < n; i += 256) {
        s += sqrtf(1e-6f + min1[i]);
        s += sqrtf(1e-6f + min2[i]);
    }
    red[threadIdx.x] = s;
    __syncthreads();
    for (int w = 128; w > 0; w >>= 1) {
        if (threadIdx.x < w) red[threadIdx.x] += red[threadIdx.x + w];
        __syncthreads();
    }
    if (threadIdx.x == 0) out[0] = red[0] * inv;
}

extern "C" void kernel_launch(void* const* d_in, const int* in_sizes, int n_in,
                              void* d_out, int out_size, void* d_ws, size_t ws_size,
                              hipStream_t stream) {
    const float* x = (const float*)d_in[0];   // (4, 4096, 3) f32
    const float* y = (const float*)d_in[1];   // (4, 4096, 3) f32
    float* out = (float*)d_out;               // scalar f32

    char* ws = (char*)d_ws;
    const int n = BS * NPTS;                                  // 16384 points
    float4* x_col = (float4*)(ws + 0 * 262144);               // 256 KB each
    float4* x_row = (float4*)(ws + 1 * 262144);
    float4* y_col = (float4*)(ws + 2 * 262144);
    float4* y_row = (float4*)(ws + 3 * 262144);
    float*  nx    = (float*)(ws + 4 * 262144);                // 64 KB each
    float*  ny    = (float*)(ws + 4 * 262144 + 65536);
    float*  min1  = (float*)(ws + 4 * 262144 + 2 * 65536);    // per x point
    float*  min2  = (float*)(ws + 4 * 262144 + 3 * 65536);    // per y point

    chamfer_setup<<<n / 256, 256, 0, stream>>>(x, y, x_col, x_row, nx,
                                               y_col, y_row, ny, n);
    // pass A: rows = y, cols = x  -> min over x per y point (min2)
    chamfer_minrows<<<BS * 32, 256, 0, stream>>>(x_col, y_row, ny, min2);
    // pass B: rows = x, cols = y  -> min over y per x point (min1)
    chamfer_minrows<<<BS * 32, 256, 0, stream>>>(y_col, x_row, nx, min1);
    chamfer_final<<<1, 256, 0, stream>>>(min1, min2, out, n, 1.0f / (float)n);
}